// thetaRNNLayer_47631187313208
// MI455X (gfx1250) — compile-verified
//
#include <hip/hip_runtime.h>
#include <math.h>

// ---------------------------------------------------------------------------
// thetaRNNLayer for MI455X (gfx1250)
//   out[0]    = seq scan  h_s = tanh(pre0_s + W_hh h_{s-1})
//   out[th]   = tanh(pre_th + out[th-1] @ W_hh^T)   (7 parallel GEMMs)
//   pre0      = x @ W_ih^T + internal[0] + b_ih + b_hh  (WMMA GEMM)
//   pre_th    = internal[th] + b_ih + b_hh              (elementwise)
// Data movement uses gfx1250 GLOBAL_LOAD_ASYNC_TO_LDS (ASYNCcnt) so global->LDS
// staging is a DMA overlapped with WMMA, not a VGPR round-trip.
// ---------------------------------------------------------------------------

#define SEQ   2048
#define HID   1024
#define TT    8          // theta + 1
#define SA    68         // LDS row stride (floats) for GEMM tiles (conflict-free)
#define SW    1032       // LDS row stride (floats) for scan W slice

typedef float v2f __attribute__((ext_vector_type(2)));
typedef float v4f __attribute__((ext_vector_type(4)));
typedef float v8f __attribute__((ext_vector_type(8)));

// async DMA: 16 bytes global -> LDS (per active lane), tracked by ASYNCcnt
__device__ __forceinline__ void async_b128(unsigned lds_byte_off, const void* gaddr) {
    asm volatile("global_load_async_to_lds_b128 %0, %1, off"
                 :: "v"(lds_byte_off), "v"(gaddr) : "memory");
}
__device__ __forceinline__ void wait_async0() {
    asm volatile("s_wait_asynccnt 0x0" ::: "memory");
}

// ---------------------------------------------------------------------------
// WMMA f32 GEMM:  D = op( A[MxK] @ W[NxK]^T + Cin [+ biases] )
// 256 threads (8 waves), 64x64 output tile, K staged in LDS (KB=64),
// double-buffered: async DMA of tile kb+64 overlaps WMMA on tile kb.
// A frag: a[v]=A[m=lane%16, k=v+2*(lane/16)];  B frag: b[v]=W[n=lane%16][same k]
// C/D:    acc[j] -> row m=j+8*(lane/16), col n=lane%16   (ISA 7.12.2 layouts)
// ---------------------------------------------------------------------------
__global__ __launch_bounds__(256) void gemm_wmma_f32(
    const float* __restrict__ A,    // [2048 x 1024]
    const float* __restrict__ W,    // [1024 x 1024], B = W^T
    const float* __restrict__ Cin,  // [2048 x 1024]
    const float* __restrict__ bi,
    const float* __restrict__ bh,
    float* __restrict__ D,          // [2048 x 1024] (may alias Cin)
    int addBias, int applyTanh)
{
    extern __shared__ float sm[];                 // 2 x (As 64xSA + Bs 64xSA)
    const unsigned lds0 = __builtin_amdgcn_groupstaticsize();

    const int t    = threadIdx.x;
    const int lane = t & 31;
    const int w    = t >> 5;
    const int l16  = lane & 15;
    const int lh   = lane >> 4;

    const int ms0 = (int)blockIdx.x * 64;   // 32 blocks in M
    const int ns0 = (int)blockIdx.y * 64;   // 16 blocks in N
    const int mt  = (w & 3) * 16;
    const int nt0 = (w >> 2) * 32;

    v8f acc0 = {};
    v8f acc1 = {};

    // issue async DMA of one 64x64 A-tile + 64x64 W-tile into LDS buffer `buf`
    auto stage = [&](int buf, int kb) {
        const unsigned base = lds0 + (unsigned)buf * (2u * 64u * SA * 4u);
        #pragma unroll
        for (int q = 0; q < 4; ++q) {
            int idx = q * 256 + t;
            int row = idx >> 4;
            int c4  = (idx & 15) * 4;
            unsigned off = base + (unsigned)(row * SA + c4) * 4u;
            async_b128(off,                  &A[(ms0 + row) * HID + kb + c4]);
            async_b128(off + 64u * SA * 4u,  &W[(ns0 + row) * HID + kb + c4]);
        }
    };

    stage(0, 0);
    wait_async0();
    __syncthreads();

    int buf = 0;
    for (int kb = 0; kb < HID; kb += 64, buf ^= 1) {
        if (kb + 64 < HID) stage(buf ^ 1, kb + 64);   // DMA next tile during WMMA

        const float* As = sm + buf * (2 * 64 * SA);
        const float* Bs = As + 64 * SA;

        #pragma unroll
        for (int k = 0; k < 64; k += 4) {
            int ko = k + 2 * lh;
            v2f a  = *(const v2f*)&As[(mt  + l16) * SA + ko];
            v2f b0 = *(const v2f*)&Bs[(nt0 + l16) * SA + ko];
            v2f b1 = *(const v2f*)&Bs[(nt0 + 16 + l16) * SA + ko];
            acc0 = __builtin_amdgcn_wmma_f32_16x16x4_f32(
                       false, a, false, b0, (short)0, acc0, false, false);
            acc1 = __builtin_amdgcn_wmma_f32_16x16x4_f32(
                       false, a, false, b1, (short)0, acc1, false, false);
        }

        wait_async0();          // next tile's DMA (issued before compute) done
        __syncthreads();        // all waves done reading current buffer
    }

    // ---- epilogue: add Cin (+biases), optional tanh ----
    #pragma unroll
    for (int j = 0; j < 8; ++j) {
        int row = ms0 + mt + j + 8 * lh;
        int c0  = ns0 + nt0 + l16;
        int c1  = c0 + 16;
        float v0 = acc0[j] + Cin[row * HID + c0];
        float v1 = acc1[j] + Cin[row * HID + c1];
        if (addBias)   { v0 += bi[c0] + bh[c0]; v1 += bi[c1] + bh[c1]; }
        if (applyTanh) { v0 = tanhf(v0);        v1 = tanhf(v1); }
        D[row * HID + c0] = v0;
        D[row * HID + c1] = v1;
    }
}

// ---------------------------------------------------------------------------
// pre[t] = internal[t] + b_ih + b_hh for t = 1..7  (elementwise, float4)
// ---------------------------------------------------------------------------
__global__ __launch_bounds__(256) void pre_init(
    const float* __restrict__ internal,
    const float* __restrict__ bi,
    const float* __restrict__ bh,
    float* __restrict__ D)
{
    const long base4 = (long)SEQ * HID / 4;      // skip t=0 slice
    long idx  = (long)blockIdx.x * 256 + threadIdx.x;
    int  hcol = (int)(idx & (HID / 4 - 1)) * 4;
    v4f v = *((const v4f*)internal + base4 + idx);
    v.x += bi[hcol + 0] + bh[hcol + 0];
    v.y += bi[hcol + 1] + bh[hcol + 1];
    v.z += bi[hcol + 2] + bh[hcol + 2];
    v.w += bi[hcol + 3] + bh[hcol + 3];
    *((v4f*)D + base4 + idx) = v;
}

// ---------------------------------------------------------------------------
// scan_init: copy initial state into global h buffer, zero barrier words
// (runs every launch -> barrier state deterministic across graph replays)
// ---------------------------------------------------------------------------
__global__ __launch_bounds__(256) void scan_init(
    const float* __restrict__ state, float* __restrict__ hg,
    unsigned* __restrict__ bar)
{
    int t = threadIdx.x;
    *((v4f*)hg + t) = *((const v4f*)state + t);
    if (t < 2) bar[t] = 0u;
}

// ---------------------------------------------------------------------------
// seq_scan: h_s = tanh(pre0_s + W_hh h_{s-1}); writes h_s over pre0 in d_out[0]
// 32 grid-resident blocks; block b holds W_hh rows [32b, 32b+32) in LDS (128KB
// fp32, async-DMA'd once) so the 2048-step recurrence streams W from LDS.
// 256 thr: output j = t>>3 (32 per block), part = t&7 (8-lane partial dots).
// Cross-block sync: sense-reversing barrier on device-scope atomics in d_ws;
// per-step 4KB h refill is an async global->LDS DMA on the critical path.
// ---------------------------------------------------------------------------
__global__ __launch_bounds__(256) void seq_scan(
    const float* __restrict__ Whh,
    float* __restrict__ base0,      // d_out[0] : pre0 in, h out
    float* __restrict__ hT,         // d_out tail (1024 floats)
    float* __restrict__ hg,         // global h buffer (d_ws)
    unsigned* __restrict__ bar)
{
    extern __shared__ float sm[];
    const unsigned lds0 = __builtin_amdgcn_groupstaticsize();
    const float* Ws = sm;                 // 32 x SW
    const float* hb = sm + 32 * SW;       // 1024
    const unsigned hb_off = lds0 + 32u * SW * 4u;

    const int t    = threadIdx.x;
    const int b    = blockIdx.x;    // 32 blocks
    const int j    = t >> 3;
    const int part = t & 7;
    const int jg   = b * 32 + j;

    // async DMA this block's 32 W_hh rows + initial h into LDS
    for (int r = 0; r < 32; ++r)
        async_b128(lds0 + (unsigned)(r * SW + t * 4) * 4u,
                   &Whh[(b * 32 + r) * HID + t * 4]);
    async_b128(hb_off + (unsigned)t * 16u, &hg[t * 4]);
    wait_async0();
    __syncthreads();

    unsigned* cnt   = bar;
    unsigned* sense = bar + 1;

    for (int s = 0; s < SEQ; ++s) {
        // partial dot over i = part, part+8, ... (Ws stride SW => conflict-free)
        float acc = 0.f;
        const float* wr = &Ws[j * SW + part];
        const float* hr = &hb[part];
        #pragma unroll 8
        for (int i = 0; i < 128; ++i)
            acc += wr[8 * i] * hr[8 * i];
        acc += __shfl_xor(acc, 1);
        acc += __shfl_xor(acc, 2);
        acc += __shfl_xor(acc, 4);

        if (part == 0) {
            float h = tanhf(base0[s * HID + jg] + acc);
            base0[s * HID + jg] = h;
            hg[jg] = h;
            if (s == SEQ - 1) hT[jg] = h;
        }
        __threadfence();        // publish this block's hg slice (device scope)
        __syncthreads();

        // ---- sense-reversing grid barrier over 32 resident blocks ----
        if (t == 0) {
            unsigned my = __hip_atomic_load(sense, __ATOMIC_RELAXED,
                                            __HIP_MEMORY_SCOPE_AGENT);
            if (atomicAdd(cnt, 1u) == 31u) {
                *cnt = 0u;
                __threadfence();
                atomicAdd(sense, 1u);                 // release all blocks
            } else {
                while (__hip_atomic_load(sense, __ATOMIC_ACQUIRE,
                                         __HIP_MEMORY_SCOPE_AGENT) == my)
                    __builtin_amdgcn_s_sleep(2);
            }
        }
        __syncthreads();
        __builtin_amdgcn_fence(__ATOMIC_ACQUIRE, "agent");  // inv near caches

        // async refill of the full updated h vector (4KB) into LDS
        async_b128(hb_off + (unsigned)t * 16u, &hg[t * 4]);
        wait_async0();
        __syncthreads();
    }
}

// ---------------------------------------------------------------------------
extern "C" void kernel_launch(void* const* d_in, const int* in_sizes, int n_in,
                              void* d_out, int out_size, void* d_ws, size_t ws_size,
                              hipStream_t stream)
{
    const float* x        = (const float*)d_in[0];   // [1,2048,1024]
    const float* internal = (const float*)d_in[1];   // [8,2048,1024]
    const float* state    = (const float*)d_in[2];   // [1,1,1024]
    const float* W_ih     = (const float*)d_in[3];   // [1024,1024]
    const float* W_hh     = (const float*)d_in[4];   // [1024,1024]
    const float* b_ih     = (const float*)d_in[5];   // [1024]
    const float* b_hh     = (const float*)d_in[6];   // [1024]

    float*    out = (float*)d_out;                   // [8,2048,1024] ++ [1024]
    float*    hg  = (float*)d_ws;                    // 1024 floats
    unsigned* bar = (unsigned*)((char*)d_ws + 4096); // 2 words

    const size_t SH       = (size_t)SEQ * HID;
    const size_t GEMM_LDS = (size_t)4 * 64 * SA * sizeof(float);        // ~68 KB (2 buffers)
    const size_t SCAN_LDS = (size_t)(32 * SW + HID) * sizeof(float);    // ~133 KB

    dim3 ggrid(SEQ / 64, HID / 64);   // 32 x 16

    // pre0 = x @ W_ih^T + internal[0] + b_ih + b_hh   -> d_out[0]
    gemm_wmma_f32<<<ggrid, 256, GEMM_LDS, stream>>>(
        x, W_ih, internal, b_ih, b_hh, out, /*addBias=*/1, /*tanh=*/0);

    // pre[1..7] = internal[1..7] + biases             -> d_out[1..7]
    pre_init<<<(unsigned)(7 * SH / 4 / 256), 256, 0, stream>>>(
        internal, b_ih, b_hh, out);

    // sequential base chain                            -> overwrites d_out[0]
    scan_init<<<1, 256, 0, stream>>>(state, hg, bar);
    seq_scan<<<32, 256, SCAN_LDS, stream>>>(
        W_hh, out, out + (size_t)TT * SH, hg, bar);

    // theta chain: out[th] = tanh(pre[th] + out[th-1] @ W_hh^T)
    for (int th = 1; th < TT; ++th) {
        gemm_wmma_f32<<<ggrid, 256, GEMM_LDS, stream>>>(
            out + (size_t)(th - 1) * SH, W_hh,
            out + (size_t)th * SH, b_ih, b_hh,
            out + (size_t)th * SH, /*addBias=*/0, /*tanh=*/1);
    }
}